// LocallyDirected1D_20418274525767
// MI455X (gfx1250) — compile-verified
//
#include <hip/hip_runtime.h>
#include <cstdint>
#include <math.h>

#ifndef __has_builtin
#define __has_builtin(x) 0
#endif

// CDNA5 async global->LDS data movement (ASYNCcnt-tracked). Guarded so the
// host pass and non-gfx1250 fall back to plain loads.
#if defined(__gfx1250__) && __has_builtin(__builtin_amdgcn_global_load_async_to_lds_b32)
#define USE_ASYNC_LDS 1
#else
#define USE_ASYNC_LDS 0
#endif

#define TO      8     // outputs per block (edge ranges are adjacent: out_idx sorted)
#define SEG_CAP 2048  // max staged edges per block (avg ~400, max ~500 for this input)

#if USE_ASYNC_LDS
// src: global memory (AS1 int*), dst: LDS (AS3 int*).
__device__ __forceinline__ void async_copy_b32(const void* gsrc, void* ldst)
{
    __builtin_amdgcn_global_load_async_to_lds_b32(
        (__attribute__((address_space(1))) int*)(uintptr_t)gsrc,
        (__attribute__((address_space(3))) int*)ldst,
        0, 0);
}
__device__ __forceinline__ void async_wait0()
{
#if __has_builtin(__builtin_amdgcn_s_wait_asynccnt)
    __builtin_amdgcn_s_wait_asynccnt(0);
#else
    asm volatile("s_wait_asynccnt 0" ::: "memory");
#endif
}
#endif

// Pass 2 (placed first so the disasm snippet shows it): one block per tile of
// TO consecutive outputs; thread t walks batch rows. The tile's edge range
// [seg_start[o0], seg_start[o0+no]) is contiguous; stage its weights + input
// indices into LDS via async global->LDS b32 loads, then each thread streams its x
// row slice once (float4 fast path when the gather is contiguous, which it is
// for in_idx = arange).
__global__ void __launch_bounds__(64)
ld1d_seg_apply(const float* __restrict__ x,
               const float* __restrict__ w,
               const float* __restrict__ bias,
               const int* __restrict__ in_idx,
               const int* __restrict__ seg_start,
               float* __restrict__ out,
               int batch, int n_in, int n_out)
{
    __shared__ float lds_w[SEG_CAP];
    __shared__ int   lds_i[SEG_CAP];
    __shared__ int   sb[TO + 1];
    __shared__ float sbias[TO];
    __shared__ int   lds_contig;

    const int o0 = blockIdx.x * TO;
    const int no = (n_out - o0 < TO) ? (n_out - o0) : TO;
    const int t  = threadIdx.x;
    const int nt = blockDim.x;

    if (t <= no) sb[t] = seg_start[o0 + t];
    if (t < no)  sbias[t] = bias[o0 + t];
    if (t == 0)  lds_contig = 1;
    __syncthreads();

    const int s = sb[0];
    const int L = sb[no] - s;

    if (L <= SEG_CAP) {
        for (int i = t; i < L; i += nt) {
#if USE_ASYNC_LDS
            async_copy_b32(w + s + i, &lds_w[i]);
            async_copy_b32(in_idx + s + i, &lds_i[i]);
#else
            lds_w[i] = w[s + i];
            lds_i[i] = in_idx[s + i];
#endif
        }
#if USE_ASYNC_LDS
        async_wait0();
#endif
        __syncthreads();

        const int base0 = (L > 0) ? lds_i[0] : 0;
        bool mine = true;
        for (int i = t; i < L; i += nt) mine = mine && (lds_i[i] == base0 + i);
        if (!mine) lds_contig = 0;   // benign same-value race
        __syncthreads();
        const bool contig = (L > 0) && (lds_contig != 0);

        for (int b = t; b < batch; b += nt) {
            const float* xrow = x + (size_t)b * (size_t)n_in;
            float* orow = out + (size_t)b * (size_t)n_out + o0;
            if (contig) {
                const float* xs = xrow + base0;   // x for local edge index 0
                for (int r = 0; r < no; ++r) {
                    int j = sb[r] - s;
                    const int je = sb[r + 1] - s;
                    float acc = 0.0f;
                    while (j < je && (reinterpret_cast<uintptr_t>(xs + j) & 15u)) {
                        acc += xs[j] * lds_w[j]; ++j;
                    }
                    for (; j + 3 < je; j += 4) {
                        const float4 xv = *reinterpret_cast<const float4*>(xs + j);
                        acc += xv.x * lds_w[j]
                             + xv.y * lds_w[j + 1]
                             + xv.z * lds_w[j + 2]
                             + xv.w * lds_w[j + 3];
                    }
                    for (; j < je; ++j) acc += xs[j] * lds_w[j];
                    orow[r] = tanhf(acc + sbias[r]);
                }
            } else {
                for (int r = 0; r < no; ++r) {
                    float acc = 0.0f;
                    for (int j = sb[r] - s; j < sb[r + 1] - s; ++j)
                        acc += xrow[lds_i[j]] * lds_w[j];
                    orow[r] = tanhf(acc + sbias[r]);
                }
            }
        }
    } else {
        // Oversized-tile fallback: stream straight from global memory.
        for (int b = t; b < batch; b += nt) {
            const float* xrow = x + (size_t)b * (size_t)n_in;
            float* orow = out + (size_t)b * (size_t)n_out + o0;
            for (int r = 0; r < no; ++r) {
                float acc = 0.0f;
                for (int j = sb[r]; j < sb[r + 1]; ++j)
                    acc += xrow[in_idx[j]] * w[j];
                orow[r] = tanhf(acc + sbias[r]);
            }
        }
    }
}

// Pass 1: segment boundaries. out_idx is sorted, so edges of output o are
// [seg_start[o], seg_start[o+1]).  seg_start has n_out+1 entries.
__global__ void ld1d_seg_bounds(const int* __restrict__ out_idx,
                                int* __restrict__ seg_start,
                                int n_out, int nnz)
{
    int o = blockIdx.x * blockDim.x + threadIdx.x;
    if (o > n_out) return;
    int lo = 0, hi = nnz;
    while (lo < hi) {
        int m = (lo + hi) >> 1;
        if (out_idx[m] < o) lo = m + 1; else hi = m;
    }
    seg_start[o] = lo;
}

extern "C" void kernel_launch(void* const* d_in, const int* in_sizes, int n_in_args,
                              void* d_out, int out_size, void* d_ws, size_t ws_size,
                              hipStream_t stream)
{
    (void)n_in_args; (void)ws_size;
    const float* x       = (const float*)d_in[0];
    const float* w       = (const float*)d_in[1];
    const float* bias    = (const float*)d_in[2];
    const int*   in_idx  = (const int*)d_in[3];
    const int*   out_idx = (const int*)d_in[4];
    // d_in[5] is n_out on device; derive all shapes host-side instead.
    const int nnz   = in_sizes[1];            // kernel: (NNZ,)
    const int n_out = in_sizes[2];            // bias: (N_OUT, 1)
    const int batch = out_size / n_out;       // out: (B, N_OUT, 1)
    const int n_in  = in_sizes[0] / batch;    // x: (B, N_IN, 1)

    int*   seg_start = (int*)d_ws;            // (n_out + 1) ints, rewritten each call
    float* out       = (float*)d_out;

    {
        const int tpb = 256;
        const int blocks = (n_out + 1 + tpb - 1) / tpb;
        ld1d_seg_bounds<<<blocks, tpb, 0, stream>>>(out_idx, seg_start, n_out, nnz);
    }
    {
        const int blocks = (n_out + TO - 1) / TO;
        ld1d_seg_apply<<<blocks, 64, 0, stream>>>(x, w, bias, in_idx, seg_start,
                                                  out, batch, n_in, n_out);
    }
}